// PersistentHomologyLoss_72318659330418
// MI455X (gfx1250) — compile-verified
//
#include <hip/hip_runtime.h>
#include <hip/hip_bf16.h>
#include <stdint.h>

#define NPIX 4096
#define HDIM 64
#define NUM_BLOCKS 12   // B(4) * classes {1,2,3}

// Path-compressed find over a u16 parent array (single-thread use).
__device__ inline int uf_find(unsigned short* parent, int a) {
    int root = a;
    while (parent[root] != root) root = parent[root];
    while (parent[a] != root) {
        int nxt = parent[a];
        parent[a] = (unsigned short)root;
        a = nxt;
    }
    return root;
}

// Recover the run-space filtration value from transformed high-32 key bits.
// run 0 (sublevel): g = v ; run 1 (superlevel): bits were (0xFFFFFFFF - b(v)), g = -v.
__device__ inline float decode_g(unsigned bits, int run) {
    unsigned ob = run ? (0xFFFFFFFFu - bits) : bits;
    float v = __uint_as_float(ob);
    return run ? -v : v;
}

__global__ __launch_bounds__(256)
void ph_loss_per_image_kernel(const float* __restrict__ x, float* __restrict__ partial) {
    __shared__ __align__(16) unsigned long long keys[NPIX];   // 32 KB (sort keys; bytes 16K..32K double as float stage)
    __shared__ unsigned short parent[NPIX];                   // 8 KB
    __shared__ unsigned short creato[NPIX];                   // 8 KB  (creator as sorted rank)
    __shared__ float  res_sum[2];
    __shared__ float  res_max[2];
    __shared__ int    res_cnt[2];
    __shared__ float  res_min;

    const int tid = threadIdx.x;
    const int bid = blockIdx.x;
    const int b   = bid / 3;
    const int c   = 1 + bid % 3;
    const float* src = x + (size_t)(b * 4 + c) * NPIX;

    float* stage = reinterpret_cast<float*>(keys) + NPIX; // bytes [16384, 32768) of keys[]

    const int dr[8] = {-1, 1, 0, 0, -1, -1, 1, 1};
    const int dc[8] = { 0, 0,-1, 1, -1,  1,-1, 1};

    for (int run = 0; run < 2; ++run) {
        // ---- Stage image into LDS with CDNA5 async global->LDS copies (ASYNCcnt) ----
        for (int i = tid * 4; i < NPIX; i += 256 * 4) {
            unsigned lds_off = (unsigned)(uintptr_t)(stage + i);
            unsigned long long gaddr = (unsigned long long)(uintptr_t)(src + i);
            asm volatile("global_load_async_to_lds_b128 %0, %1, off"
                         :: "v"(lds_off), "v"(gaddr) : "memory");
        }
        asm volatile("s_wait_asynccnt 0" ::: "memory");
        __syncthreads();

        // ---- Pull my 16 values into registers (stage aliases upper half of keys[]) ----
        float vloc[16];
#pragma unroll
        for (int e = 0; e < 16; ++e) vloc[e] = stage[tid + 256 * e];
        __syncthreads();

        // ---- Build sort keys + init union-find parent ----
#pragma unroll
        for (int e = 0; e < 16; ++e) {
            int i = tid + 256 * e;
            unsigned bits = __float_as_uint(vloc[e]);          // v in [0,1): bits monotone in v
            if (run) bits = 0xFFFFFFFFu - bits;                // monotone in -v, stable ties by idx
            keys[i] = ((unsigned long long)bits << 32) | (unsigned)i;
        }
        for (int i = tid; i < NPIX; i += 256) parent[i] = 0xFFFFu;
        __syncthreads();

        // ---- Bitonic sort of 4096 u64 keys in LDS ----
        for (unsigned k = 2; k <= NPIX; k <<= 1) {
            for (unsigned j = k >> 1; j > 0; j >>= 1) {
                for (unsigned i = tid; i < NPIX; i += 256) {
                    unsigned ixj = i ^ j;
                    if (ixj > i) {
                        unsigned long long a = keys[i];
                        unsigned long long bb = keys[ixj];
                        bool up = ((i & k) == 0);
                        if ((a > bb) == up) { keys[i] = bb; keys[ixj] = a; }
                    }
                }
                __syncthreads();
            }
        }

        // ---- Serial elder-rule union-find over sorted order (thread 0) ----
        if (tid == 0) {
            float sumsq = 0.0f, mx = 0.0f;
            int cnt = 0;
            const int nnb = run ? 4 : 8;   // H0: 8-conn sublevel; H1 dual: 4-conn superlevel
            if (run == 0) res_min = __uint_as_float((unsigned)(keys[0] >> 32));
            for (int pos = 0; pos < NPIX; ++pos) {
                unsigned long long kk = keys[pos];
                int idx = (int)(unsigned)kk;
                unsigned curbits = (unsigned)(kk >> 32);
                parent[idx] = (unsigned short)idx;
                creato[idx] = (unsigned short)pos;
                int r = idx >> 6, cc = idx & 63;
                for (int o = 0; o < nnb; ++o) {
                    int rr = r + dr[o], c2 = cc + dc[o];
                    if (rr < 0 || rr >= HDIM || c2 < 0 || c2 >= HDIM) continue;
                    int nb = (rr << 6) + c2;
                    if (parent[nb] == 0xFFFFu) continue;
                    int ra = uf_find(parent, idx);
                    int rb = uf_find(parent, nb);
                    if (ra == rb) continue;
                    int ca = creato[ra], cb = creato[rb];
                    int elder, young;
                    if (ca < cb) { elder = ra; young = rb; }
                    else         { elder = rb; young = ra; }
                    int ybirth = creato[young];
                    unsigned bbits = (unsigned)(keys[ybirth] >> 32);
                    if (curbits > bbits) {          // strictly positive persistence
                        float pers = decode_g(curbits, run) - decode_g(bbits, run);
                        sumsq += pers * pers;
                        if (pers > mx) mx = pers;
                        ++cnt;
                    }
                    parent[young] = (unsigned short)elder;
                }
            }
            res_sum[run] = sumsq;
            res_max[run] = mx;
            res_cnt[run] = cnt;
        }
        __syncthreads();
    }

    if (tid == 0) {
        // TOPO: class1 {h0:1,h1:0}, class2 {h0:1,h1:1}, class3 {h0:2,h1:0}
        int nf0 = (c == 3) ? 2 : 1;
        int nf1 = (c == 2) ? 1 : 0;
        float minv = res_min;
        float S0 = res_sum[0], M0 = res_max[0]; int n0 = res_cnt[0];
        float S1 = res_sum[1], M1 = res_max[1]; int n1 = res_cnt[1];

        float loss0 = minv * minv + S0;
        if (nf0 == 2) loss0 += (n0 >= 1) ? ((1.0f - M0) * (1.0f - M0) - M0 * M0) : 1.0f;

        float loss1;
        if (nf1 == 1) loss1 = (n1 >= 1) ? ((1.0f - M1) * (1.0f - M1) + S1 - M1 * M1) : 1.0f;
        else          loss1 = S1;

        partial[bid] = loss0 + loss1;
    }
}

__global__ void ph_loss_reduce_kernel(const float* __restrict__ partial, float* __restrict__ out) {
    // Single thread, fixed summation order -> deterministic.
    float s = 0.0f;
    for (int i = 0; i < NUM_BLOCKS; ++i) s += partial[i];
    out[0] = s * 0.25f;   // / B
}

extern "C" void kernel_launch(void* const* d_in, const int* in_sizes, int n_in,
                              void* d_out, int out_size, void* d_ws, size_t ws_size,
                              hipStream_t stream) {
    (void)in_sizes; (void)n_in; (void)out_size; (void)ws_size;
    const float* x = (const float*)d_in[0];       // (1,4,4,64,64) float32
    float* partial = (float*)d_ws;                // 12 floats of scratch
    float* out = (float*)d_out;                   // scalar float32 loss

    ph_loss_per_image_kernel<<<NUM_BLOCKS, 256, 0, stream>>>(x, partial);
    ph_loss_reduce_kernel<<<1, 1, 0, stream>>>(partial, out);
}